// TorchSABR_LMM_39513699123290
// MI455X (gfx1250) — compile-verified
//
#include <hip/hip_runtime.h>
#include <hip/hip_bf16.h>

typedef float v2f  __attribute__((ext_vector_type(2)));
typedef float v8f  __attribute__((ext_vector_type(8)));
typedef unsigned int u32x4 __attribute__((ext_vector_type(4)));
typedef int   i32x4 __attribute__((ext_vector_type(4)));
typedef int   i32x8 __attribute__((ext_vector_type(8)));

#define NN        40
#define MSTEPS    100
#define NPATHS    8192
#define FOURN     160
#define RHOX      0.8f
#define PPB       16      // paths per block (WMMA M dimension)
#define TPB       128     // 4 wave32
#define ZROW      180     // TDM-padded Z row stride: 160 + 4 DW pad per 32 DW (even + conflict-free)
#define DWSTR     164     // dW tile row stride
#define GSTR      44      // gamma tile row stride (A-frag conflict-free, 8B aligned)
#define SSTR      48      // s tile row stride (N padded 40->48)

// ---- LDS layout (float offsets) ----
#define OFF_LP    0                          // L packed in B-frag order [kc][j][khalf] float2: 25600
#define OFF_ZB0   25600                      // Z tile double buffer, 16 x ZROW each
#define OFF_ZB1   (OFF_ZB0 + PPB*ZROW)
#define OFF_DWB   (OFF_ZB1 + PPB*ZROW)       // dW tile 16 x DWSTR
#define OFF_FO    (OFF_DWB + PPB*DWSTR)      // state arrays, 640 each
#define OFF_FE    (OFF_FO + PPB*NN)
#define OFF_VO    (OFF_FE + PPB*NN)
#define OFF_VE    (OFF_VO + PPB*NN)
#define OFF_GS    (OFF_VE + PPB*NN)          // gamma tile 16 x GSTR
#define OFF_SS    (OFF_GS + PPB*GSTR)        // s tile 16 x SSTR
#define OFF_DCB   (OFF_SS + PPB*SSTR)        // drift_corr^T packed B-frags: 10 kc x 48 j x 2 = 1920
#define OFF_TN    (OFF_DCB + 1920)
#define OFF_TAU   (OFF_TN + NN)
#define OFF_NU    (OFF_TAU + NN)
#define OFF_VDC   (OFF_NU + NN)
#define SMEM_FLOATS (OFF_VDC + NN)           // 40096 floats = 160384 B -> 2 WGs / 320KB WGP

// ---- Tensor Data Mover: DMA one 16-path x 160-float Z tile into LDS (ISA ch.8 D#) ----
__device__ __forceinline__ void tdm_load_z_tile(const float* gptr, unsigned lds_byte_off)
{
    unsigned long long ga = (unsigned long long)(uintptr_t)gptr;
    u32x4 g0;
    g0.x = 1u;                                   // count=1, user descriptor, no gather
    g0.y = lds_byte_off;                         // LDS byte address of tile start
    g0.z = (unsigned)(ga & 0xffffffffu);         // global_addr[31:0]
    g0.w = (unsigned)(ga >> 32) | (2u << 30);    // global_addr[56:32] | type=2 ("image")

    i32x8 g1;
    g1[0] = (2 << 16)          // data_size = 4 bytes
          | (1 << 20)          // pad_enable (LDS-side padding on load)
          | (4 << 22)          // pad_interval: every 32 DWORDs
          | (3 << 25);         // pad_amount: 4 DWORDs  -> row stride 160+20 = 180 floats
    g1[1] = (FOURN << 16);     // tensor_dim0 = 160
    g1[2] = (PPB   << 16);     // tensor_dim1 = 16
    g1[3] = (FOURN << 16);     // tile_dim0 = 160
    g1[4] = PPB;               // tile_dim1 = 16, tile_dim2 = 0
    g1[5] = MSTEPS * FOURN;    // tensor_dim0_stride = 16000 elements (path-to-path)
    g1[6] = 0;
    g1[7] = 0;

    i32x4 gz = {0, 0, 0, 0};   // groups 2/3 unused (2-D tensor)
#if defined(__clang_major__) && (__clang_major__ >= 23)
    i32x8 gz8 = {0, 0, 0, 0, 0, 0, 0, 0};
    __builtin_amdgcn_tensor_load_to_lds(g0, g1, gz, gz, gz8, 0);
#else
    __builtin_amdgcn_tensor_load_to_lds(g0, g1, gz, gz, 0);
#endif
}

extern "C" __global__ __launch_bounds__(TPB)
void sabr_lmm_fused(const float* __restrict__ Z,   const float* __restrict__ T,
                    const float* __restrict__ tau, const float* __restrict__ F0o,
                    const float* __restrict__ F0e, const float* __restrict__ alphas,
                    const float* __restrict__ nus, const float* __restrict__ L,
                    const float* __restrict__ dc,  float* __restrict__ out)
{
    extern __shared__ float sm[];
    float* Lp   = sm + OFF_LP;
    float* dWb  = sm + OFF_DWB;
    float* foS  = sm + OFF_FO;
    float* feS  = sm + OFF_FE;
    float* voS  = sm + OFF_VO;
    float* veS  = sm + OFF_VE;
    float* gS   = sm + OFF_GS;
    float* sS   = sm + OFF_SS;
    float* dcB  = sm + OFF_DCB;
    float* TnS  = sm + OFF_TN;
    float* tauS = sm + OFF_TAU;
    float* nuS  = sm + OFF_NU;
    float* vdcS = sm + OFF_VDC;

    const int tid = threadIdx.x;
    const int p0  = blockIdx.x * PPB;
    const int wv   = tid >> 5;
    const int lane = tid & 31;
    const int nL   = lane & 15;
    const int kh   = lane >> 4;

    // kick off step-0 Z tile DMA immediately (overlaps all of init)
    if (wv == 0)
        tdm_load_z_tile(&Z[(size_t)p0 * MSTEPS * FOURN], OFF_ZB0 * 4u);

    const float Tend = T[NN];
    const float dt   = Tend / (float)MSTEPS;
    const float sqdt = sqrtf(dt);

    float* out_time = out;
    float* out_ois  = out + (MSTEPS + 1);
    float* out_eur  = out_ois + (size_t)NPATHS * (MSTEPS + 1) * NN;

    // ---- init: pack L and drift_corr^T into WMMA B-fragment order, constants, state ----
    for (int e = tid; e < (FOURN * FOURN) / 2; e += TPB) {   // 12800 float2
        int kc = e / 320;
        int r  = e - kc * 320;
        int j  = r >> 1;
        int khf = r & 1;
        int k0 = kc * 4 + 2 * khf;
        v2f v;
        v.x = L[j * FOURN + k0];
        v.y = L[j * FOURN + k0 + 1];
        *(v2f*)(Lp + 2 * e) = v;
    }
    for (int e = tid; e < 10 * SSTR * 2 / 2; e += TPB) {     // 480 float2: dcB[kc][j<48][khalf]
        int kc = e / (SSTR * 2 / 2);                         // 96 float2 per kc
        int r  = e - kc * 96;
        int j  = r >> 1;
        int khf = r & 1;
        int k0 = kc * 4 + 2 * khf;                           // k0 <= 38, k0+1 <= 39
        v2f v;
        v.x = (j < NN) ? dc[j * NN + k0]     : 0.0f;         // B[k][j] = drift_corr[j][k]
        v.y = (j < NN) ? dc[j * NN + k0 + 1] : 0.0f;
        *(v2f*)(dcB + 2 * e) = v;
    }
    if (tid < NN) {
        TnS[tid]  = T[tid];
        tauS[tid] = tau[tid];
        float nu  = nus[tid];
        nuS[tid]  = nu;
        vdcS[tid] = -0.5f * nu * nu * dt;
    }
    for (int item = tid; item < PPB * NN; item += TPB) {
        int pl = item / NN, i = item - pl * NN;
        float fo = F0o[i], fe = F0e[i], al = alphas[i];
        foS[item] = fo; feS[item] = fe; voS[item] = al; veS[item] = al;
        size_t ob = ((size_t)(p0 + pl) * (MSTEPS + 1)) * NN + i;   // m = 0 row
        out_ois[ob] = fo;
        out_eur[ob] = fe;
    }
    if (blockIdx.x == 0 && tid <= MSTEPS)
        out_time[tid] = Tend * (float)tid / (float)MSTEPS;
    __syncthreads();

    for (int m = 0; m < MSTEPS; ++m) {
        const float t_curr = Tend * (float)m / (float)MSTEPS;

        // current Z buffer is ready once wave0's TENSORcnt drains; then broadcast via barrier
        __builtin_amdgcn_s_wait_tensorcnt(0);
        __syncthreads();
        // prefetch next step's tile into the other buffer via TDM (overlaps everything below)
        if (wv == 0 && (m + 1) < MSTEPS)
            tdm_load_z_tile(&Z[((size_t)p0 * MSTEPS + (m + 1)) * FOURN],
                            (((m + 1) & 1) ? OFF_ZB1 : OFF_ZB0) * 4u);

        const float* zb = sm + ((m & 1) ? OFF_ZB1 : OFF_ZB0);

        // ---- dW = (Z_tile @ L^T) * sqrt(dt) via fp32 WMMA 16x16x4 ----
        for (int jt = wv; jt < FOURN / 16; jt += 4) {
            v8f acc = {};
            #pragma unroll 4
            for (int kc = 0; kc < FOURN / 4; ++kc) {
                int k = kc * 4 + 2 * kh;
                v2f a = *(const v2f*)(zb + nL * ZROW + k + 4 * (k >> 5));   // TDM pad-adjusted
                v2f b = *(const v2f*)(Lp + 2 * (kc * 320 + (jt * 16 + nL) * 2 + kh));
                acc = __builtin_amdgcn_wmma_f32_16x16x4_f32(
                        false, a, false, b, (short)0, acc, false, false);
            }
            #pragma unroll
            for (int r = 0; r < 8; ++r)
                dWb[(r + 8 * kh) * DWSTR + jt * 16 + nL] = acc[r] * sqdt;
        }
        __syncthreads();

        // ---- SABR predictor-corrector; 640 (path,rate) items over 128 threads ----
        float fo_[5], fe_[5], vo_[5], ve_[5], mo_[5], me_[5], dfo_[5], dfe_[5], act_[5];

        // E1: predictor gamma -> gS
        #pragma unroll
        for (int u = 0; u < 5; ++u) {
            int item = tid + u * TPB;
            int pl = item / NN, i = item - pl * NN;
            float fo = foS[item], vo = voS[item];
            float act = (TnS[i] > t_curr) ? 1.0f : 0.0f;
            gS[pl * GSTR + i] = act * tauS[i] * vo / (1.0f + tauS[i] * fo);
            fo_[u] = fo; vo_[u] = vo;
            fe_[u] = feS[item]; ve_[u] = veS[item];
            act_[u] = act;
        }
        __syncthreads();

        // W1: s1 = gamma @ drift_corr^T via WMMA (3 j-tiles on waves 0..2)
        if (wv < 3) {
            const int jt = wv;
            v8f acc = {};
            #pragma unroll
            for (int kc = 0; kc < 10; ++kc) {
                v2f a = *(const v2f*)(gS + nL * GSTR + kc * 4 + 2 * kh);
                v2f b = *(const v2f*)(dcB + 2 * (kc * 96 + (jt * 16 + nL) * 2 + kh));
                acc = __builtin_amdgcn_wmma_f32_16x16x4_f32(
                        false, a, false, b, (short)0, acc, false, false);
            }
            #pragma unroll
            for (int r = 0; r < 8; ++r)
                sS[(r + 8 * kh) * SSTR + jt * 16 + nL] = acc[r];
        }
        __syncthreads();

        // E2: predictor drift, corrector gamma -> gS
        #pragma unroll
        for (int u = 0; u < 5; ++u) {
            int item = tid + u * TPB;
            int pl = item / NN, i = item - pl * NN;
            float s  = sS[pl * SSTR + i];
            float mo = -vo_[u] * s;
            float me = -ve_[u] * RHOX * s;
            float dfo = dWb[pl * DWSTR + i];
            float fo_p = fo_[u] + mo * dt + vo_[u] * dfo;
            gS[pl * GSTR + i] = act_[u] * tauS[i] * vo_[u] / (1.0f + tauS[i] * fo_p);
            mo_[u] = mo; me_[u] = me; dfo_[u] = dfo;
            dfe_[u] = dWb[pl * DWSTR + NN + i];
        }
        __syncthreads();

        // W2: s2 via WMMA
        if (wv < 3) {
            const int jt = wv;
            v8f acc = {};
            #pragma unroll
            for (int kc = 0; kc < 10; ++kc) {
                v2f a = *(const v2f*)(gS + nL * GSTR + kc * 4 + 2 * kh);
                v2f b = *(const v2f*)(dcB + 2 * (kc * 96 + (jt * 16 + nL) * 2 + kh));
                acc = __builtin_amdgcn_wmma_f32_16x16x4_f32(
                        false, a, false, b, (short)0, acc, false, false);
            }
            #pragma unroll
            for (int r = 0; r < 8; ++r)
                sS[(r + 8 * kh) * SSTR + jt * 16 + nL] = acc[r];
        }
        __syncthreads();

        // E3: corrector, vol update, state + history writeback
        #pragma unroll
        for (int u = 0; u < 5; ++u) {
            int item = tid + u * TPB;
            int pl = item / NN, i = item - pl * NN;
            float s2  = sS[pl * SSTR + i];
            float mop = -vo_[u] * s2;
            float mep = -ve_[u] * RHOX * s2;
            float fo_n = fo_[u] + 0.5f * (mo_[u] + mop) * dt + vo_[u] * dfo_[u];
            float fe_n = fe_[u] + 0.5f * (me_[u] + mep) * dt + ve_[u] * dfe_[u];
            float dvo = dWb[pl * DWSTR + 2 * NN + i];
            float dve = dWb[pl * DWSTR + 3 * NN + i];
            float vo_n = vo_[u] * __expf(vdcS[i] + nuS[i] * dvo);
            float ve_n = ve_[u] * __expf(vdcS[i] + nuS[i] * dve);
            bool on = act_[u] > 0.f;
            float fo2 = on ? fo_n : fo_[u];
            float fe2 = on ? fe_n : fe_[u];
            float vo2 = on ? vo_n : vo_[u];
            float ve2 = on ? ve_n : ve_[u];
            foS[item] = fo2; feS[item] = fe2; voS[item] = vo2; veS[item] = ve2;
            size_t ob = ((size_t)(p0 + pl) * (MSTEPS + 1) + (m + 1)) * NN + i;
            out_ois[ob] = fo2;
            out_eur[ob] = fe2;
        }
        __syncthreads();   // protect gS/sS/dWb before next iteration overwrites
    }
}

extern "C" void kernel_launch(void* const* d_in, const int* in_sizes, int n_in,
                              void* d_out, int out_size, void* d_ws, size_t ws_size,
                              hipStream_t stream) {
    (void)in_sizes; (void)n_in; (void)out_size; (void)d_ws; (void)ws_size;
    const float* Z    = (const float*)d_in[0];
    const float* T    = (const float*)d_in[1];
    const float* tau  = (const float*)d_in[2];
    const float* F0o  = (const float*)d_in[3];
    const float* F0e  = (const float*)d_in[4];
    const float* al   = (const float*)d_in[5];
    const float* nus  = (const float*)d_in[6];
    const float* L    = (const float*)d_in[7];
    const float* dcm  = (const float*)d_in[8];
    float* out = (float*)d_out;

    dim3 grid(NPATHS / PPB);          // 512 workgroups, 16 paths each
    dim3 block(TPB);                  // 4 wave32
    size_t smem = SMEM_FLOATS * sizeof(float);
    hipLaunchKernelGGL(sabr_lmm_fused, grid, block, smem, stream,
                       Z, T, tau, F0o, F0e, al, nus, L, dcm, out);
}